// MarketEncoder_19567871000773
// MI455X (gfx1250) — compile-verified
//
#include <hip/hip_runtime.h>
#include <math.h>

#define B_    64
#define T_    65536
#define WIN   14
#define DIM   256
#define HID   512
#define EPSF  1e-8f

#define TILE  4096
#define BLK   256
#define NT    (T_ / TILE)     /* 16 */
#define ITERS (TILE / BLK)    /* 16 */
#define CHK   17              /* per-thread scan chunk; 17 coprime with 64 banks */
#define SC    (BLK * CHK)     /* 4352 scan capacity */
#define NSTG  (TILE + WIN)    /* 4110 staged entries per tile */

typedef __attribute__((ext_vector_type(2))) float v2f;
typedef __attribute__((ext_vector_type(8))) float v8f;
typedef __attribute__((ext_vector_type(4))) unsigned int v4u;
typedef __attribute__((ext_vector_type(4))) int v4i;
typedef __attribute__((ext_vector_type(8))) int v8i;

// ---------------------------------------------------------------------------
// Dual inclusive prefix scan of p1/p2[SC] with 256 threads.
// Chunk scan (stride-17 -> LDS bank-conflict-free), Hillis-Steele over totals.
// ---------------------------------------------------------------------------
__device__ __forceinline__ void scan2(float* p1, float* p2,
                                      float* tot1, float* tot2, int tid) {
  const int base = tid * CHK;
  float r1 = 0.f, r2 = 0.f;
#pragma unroll
  for (int j = 0; j < CHK; ++j) {
    r1 += p1[base + j]; p1[base + j] = r1;
    r2 += p2[base + j]; p2[base + j] = r2;
  }
  tot1[tid] = r1; tot2[tid] = r2;
  __syncthreads();
  for (int off = 1; off < BLK; off <<= 1) {
    const float a = (tid >= off) ? tot1[tid - off] : 0.f;
    const float b = (tid >= off) ? tot2[tid - off] : 0.f;
    __syncthreads();
    tot1[tid] += a; tot2[tid] += b;
    __syncthreads();
  }
  const float o1 = (tid > 0) ? tot1[tid - 1] : 0.f;
  const float o2 = (tid > 0) ? tot2[tid - 1] : 0.f;
#pragma unroll
  for (int j = 0; j < CHK; ++j) { p1[base + j] += o1; p2[base + j] += o2; }
  __syncthreads();
}

// ---------------------------------------------------------------------------
// Kernel 1: single HBM pass. TDM DMAs the volume tile into LDS while the
// returns phase runs; rolling W=14 stats come from per-tile prefix sums.
// ---------------------------------------------------------------------------
__global__ __launch_bounds__(BLK) void k_feat_reduce(
    const float* __restrict__ close, const float* __restrict__ high,
    const float* __restrict__ low,   const float* __restrict__ open_,
    const float* __restrict__ volume, float* __restrict__ partials) {
  const int b = blockIdx.y, tile = blockIdx.x, tid = threadIdx.x;
  const size_t row = (size_t)b * T_;
  const float* c  = close  + row;
  const float* hg = high   + row;
  const float* lo = low    + row;
  const float* op = open_  + row;
  const float* vo = volume + row;
  const int t0 = tile * TILE;

  __shared__ float p1[SC];
  __shared__ float p2[SC];
  __shared__ float s_vol[NSTG];
  __shared__ float tot1[BLK];
  __shared__ float tot2[BLK];
  __shared__ float red[10][BLK];

  // ---- Issue TDM: volume[t0-WIN .. t0+TILE-1] -> s_vol (async, TENSORcnt) --
  {
    int start_t = t0 - WIN;
    int dst_off = 0;
    if (start_t < 0) { dst_off = -start_t; start_t = 0; }
    const int cnt = NSTG - dst_off;
    const unsigned long long gaddr = (unsigned long long)(size_t)(vo + start_t);
    const unsigned int laddr = (unsigned int)(size_t)(&s_vol[dst_off]);
    v4u g0;
    g0.x = 1u;                                               // count=1, no gather
    g0.y = laddr;                                            // lds_addr
    g0.z = (unsigned int)(gaddr & 0xFFFFFFFFu);              // global_addr lo
    g0.w = (unsigned int)((gaddr >> 32) & 0x1FFFFFFu) | (2u << 30); // hi | type=2
    v8i g1;
    g1[0] = 0x20000;                        // wg_mask=0, data_size=2 (4B)
    g1[1] = (int)((unsigned int)cnt << 16); // tensor_dim0[15:0]
    g1[2] = (int)(1u << 16);                // tensor_dim0 hi=0 | tensor_dim1=1
    g1[3] = (int)((unsigned int)cnt << 16); // tile_dim0 = cnt
    g1[4] = 0;                              // tile_dim1=0, tile_dim2=0
    g1[5] = cnt;                            // tensor_dim0_stride lo
    g1[6] = 0;
    g1[7] = 0;
    const v4i z4 = {0, 0, 0, 0};
    if (tid < 32) {
#if defined(__clang_major__) && (__clang_major__ >= 23)
      const v8i z8 = {0, 0, 0, 0, 0, 0, 0, 0};
      __builtin_amdgcn_tensor_load_to_lds(g0, g1, z4, z4, z8, 0);
#else
      __builtin_amdgcn_tensor_load_to_lds(g0, g1, z4, z4, 0);
#endif
    }
    if (tile == 0 && tid < WIN) s_vol[tid] = 0.f;  // pre-history pad
  }

  // Prefetch next tile of the streams we load through VGPRs.
  if (tile + 1 < NT) {
    const int pt = t0 + TILE + tid * ITERS;
    __builtin_prefetch(c  + pt, 0, 0);
    __builtin_prefetch(hg + pt, 0, 0);
    __builtin_prefetch(lo + pt, 0, 0);
    __builtin_prefetch(op + pt, 0, 0);
  }

  float a1[5] = {0.f, 0.f, 0.f, 0.f, 0.f};
  float a2[5] = {0.f, 0.f, 0.f, 0.f, 0.f};

  // ---- Phase A: returns prefix sums; 'vol' = rolling std from prefix diff --
  for (int idx = tid; idx < SC; idx += BLK) {
    float r = 0.f;
    if (idx < NSTG) {
      const int t = t0 - WIN + idx;
      if (t >= 1) { const float cp = c[t - 1]; r = (c[t] - cp) / (cp + EPSF); }
    }
    p1[idx] = r; p2[idx] = r * r;
  }
  __syncthreads();
  scan2(p1, p2, tot1, tot2, tid);

  // Feature 0 (raw returns) totals over this tile: free from the scan.
  const float f0sum = p1[WIN + TILE - 1] - p1[WIN - 1];
  const float f0sq  = p2[WIN + TILE - 1] - p2[WIN - 1];

  for (int i = 0; i < ITERS; ++i) {
    const int idx = tid + BLK * i;
    const int t   = t0 + idx;
    float volf = 0.f;
    if (t >= WIN) {
      const float s = p1[idx + WIN] - p1[idx];
      const float q = p2[idx + WIN] - p2[idx];
      const float m = s * (1.f / WIN);
      const float var = fmaxf((q - (float)WIN * m * m) * (1.f / (WIN - 1)), 0.f);
      volf = sqrtf(var);
    }
    a1[1] += volf; a2[1] = fmaf(volf, volf, a2[1]);
  }
  __syncthreads();  // done reading phase-A prefixes

  // ---- Phase B: wait for TDM, volume prefix sums, volume_z + elementwise ---
  if (tid < 32) __builtin_amdgcn_s_wait_tensorcnt(0);
  __syncthreads();

  for (int idx = tid; idx < SC; idx += BLK) {
    const float v = (idx < NSTG) ? s_vol[idx] : 0.f;
    p1[idx] = v; p2[idx] = v * v;
  }
  __syncthreads();
  scan2(p1, p2, tot1, tot2, tid);

  for (int i = 0; i < ITERS; ++i) {
    const int idx = tid + BLK * i;
    const int t   = t0 + idx;
    const float vt = s_vol[idx + WIN];
    float vz = vt;  // t < WIN: (v - 0) / 1
    if (t >= WIN) {
      const float s = p1[idx + WIN] - p1[idx];
      const float q = p2[idx + WIN] - p2[idx];
      const float m = s * (1.f / WIN);
      const float var = fmaxf((q - (float)WIN * m * m) * (1.f / (WIN - 1)), 0.f);
      vz = (vt - m) / (sqrtf(var) + EPSF);
    }
    const float ov     = op[t];
    const float range  = (hg[t] - lo[t]) / (ov + EPSF);
    const float pratio = c[t] / (ov + EPSF) - 1.f;
    a1[2] += vz;     a2[2] = fmaf(vz, vz, a2[2]);
    a1[3] += range;  a2[3] = fmaf(range, range, a2[3]);
    a1[4] += pratio; a2[4] = fmaf(pratio, pratio, a2[4]);
  }
  if (tid == 0) { a1[0] = f0sum; a2[0] = f0sq; }

  // ---- Deterministic block tree-reduction of 10 partial sums ---------------
  __syncthreads();
#pragma unroll
  for (int k = 0; k < 5; ++k) { red[k][tid] = a1[k]; red[5 + k][tid] = a2[k]; }
  __syncthreads();
  for (int off = BLK / 2; off > 0; off >>= 1) {
    if (tid < off) {
#pragma unroll
      for (int k = 0; k < 10; ++k) red[k][tid] += red[k][tid + off];
    }
    __syncthreads();
  }
  if (tid < 10) partials[((size_t)b * NT + tile) * 10 + tid] = red[tid][0];
}

// ---------------------------------------------------------------------------
// Kernel 2: fold tile partials -> per-(b,f) mean of standardized feature.
// fmean = (s1/T - mu)/sd with mu = s1/T (exact-math zero; reference's nonzero
// values are fp cancellation noise ~1e-9, well inside tolerance).
// ---------------------------------------------------------------------------
__global__ void k_fmean(const float* __restrict__ partials, float* __restrict__ fmean) {
  const int b = blockIdx.x, f = threadIdx.x;
  if (f < 5) {
    float s1 = 0.f, s2 = 0.f;
    for (int i = 0; i < NT; ++i) {
      s1 += partials[((size_t)b * NT + i) * 10 + f];
      s2 += partials[((size_t)b * NT + i) * 10 + 5 + f];
    }
    const float mu  = s1 / (float)T_;
    const float var = fmaxf((s2 - (float)T_ * mu * mu) / (float)(T_ - 1), 0.f);
    const float sd  = sqrtf(var) + EPSF;
    fmean[b * 5 + f] = (s1 / (float)T_ - mu) / sd;
  }
}

// ---------------------------------------------------------------------------
// Kernel 3a: h0[b][d] = sum_k fmean[b][k] * Wp[d][k] + bp[d]   (K=5, FMA)
// ---------------------------------------------------------------------------
__global__ __launch_bounds__(DIM) void k_proj(const float* __restrict__ fmean,
                                              const float* __restrict__ Wp,
                                              const float* __restrict__ bp,
                                              float* __restrict__ h0) {
  const int b = blockIdx.x, d = threadIdx.x;
  float acc = bp[d];
#pragma unroll
  for (int k = 0; k < 5; ++k) acc = fmaf(fmean[b * 5 + k], Wp[d * 5 + k], acc);
  h0[(size_t)b * DIM + d] = acc;
}

// ---------------------------------------------------------------------------
// WMMA fp32 GEMM: D[64 x N] = A[64 x K] * W^T + bias, optional exact GELU.
// One wave per 16x16 tile; K chained through V_WMMA_F32_16X16X4_F32.
// A 16x4 fp32: lanes 0-15 -> K=0,1 ; lanes 16-31 -> K=2,3.
// B row-striped like C/D: v0 = rows K0/K2, v1 = rows K1/K3.
// C/D: vgpr j, lane l -> row = j + (l>>4)*8, col = l&15.
// ---------------------------------------------------------------------------
template <int K, int N, bool GELU>
__global__ __launch_bounds__(128) void k_gemm_wmma(const float* __restrict__ A,
                                                   const float* __restrict__ W,
                                                   const float* __restrict__ bias,
                                                   float* __restrict__ D) {
  const int wave = blockIdx.x * (blockDim.x >> 5) + (threadIdx.x >> 5);
  const int nt = wave % (N / 16);
  const int mt = wave / (N / 16);
  const int l  = threadIdx.x & 31;
  const int hi = l >> 4;
  const int lm = l & 15;
  const float* Arow = A + (size_t)(mt * 16 + lm) * K;
  const float* Wrow = W + (size_t)(nt * 16 + lm) * K;
  const int n = nt * 16 + lm;

  v8f acc = {0.f, 0.f, 0.f, 0.f, 0.f, 0.f, 0.f, 0.f};
  for (int kb = 0; kb < K; kb += 4) {
    const int ka = kb + hi * 2;
    v2f av; av.x = Arow[ka]; av.y = Arow[ka + 1];
    v2f bv; bv.x = Wrow[ka]; bv.y = Wrow[ka + 1];
    acc = __builtin_amdgcn_wmma_f32_16x16x4_f32(
        /*neg_a=*/false, av, /*neg_b=*/false, bv,
        /*c_mod=*/(short)0, acc, /*reuse_a=*/false, /*reuse_b=*/false);
  }

  const float bn = bias[n];
#pragma unroll
  for (int j = 0; j < 8; ++j) {
    const int rowM = mt * 16 + hi * 8 + j;
    float x = acc[j] + bn;
    if (GELU) x = 0.5f * x * (1.f + erff(x * 0.7071067811865476f));
    D[(size_t)rowM * N + n] = x;
  }
}

// ---------------------------------------------------------------------------
extern "C" void kernel_launch(void* const* d_in, const int* in_sizes, int n_in,
                              void* d_out, int out_size, void* d_ws, size_t ws_size,
                              hipStream_t stream) {
  (void)in_sizes; (void)n_in; (void)out_size; (void)ws_size;
  const float* close  = (const float*)d_in[0];
  const float* high   = (const float*)d_in[1];
  const float* low    = (const float*)d_in[2];
  const float* open_  = (const float*)d_in[3];
  const float* volume = (const float*)d_in[4];
  const float* Wp = (const float*)d_in[5];
  const float* bp = (const float*)d_in[6];
  const float* W1 = (const float*)d_in[7];
  const float* b1 = (const float*)d_in[8];
  const float* W2 = (const float*)d_in[9];
  const float* b2 = (const float*)d_in[10];
  float* out = (float*)d_out;

  float* ws       = (float*)d_ws;
  float* partials = ws;                       // B_*NT*10   = 10240
  float* fmean    = partials + B_ * NT * 10;  // B_*5       = 320
  float* h0       = fmean + B_ * 5;           // B_*DIM     = 16384
  float* h1       = h0 + B_ * DIM;            // B_*HID     = 32768

  k_feat_reduce<<<dim3(NT, B_), BLK, 0, stream>>>(close, high, low, open_, volume, partials);
  k_fmean<<<dim3(B_), 32, 0, stream>>>(partials, fmean);
  k_proj<<<dim3(B_), DIM, 0, stream>>>(fmean, Wp, bp, h0);
  // h1 = gelu(h0 @ W1^T + b1): (64/16)*(512/16)=128 waves -> 32 blocks x 4 waves
  k_gemm_wmma<DIM, HID, true><<<dim3(32), 128, 0, stream>>>(h0, W1, b1, h1);
  // out = h1 @ W2^T + b2: (64/16)*(256/16)=64 waves -> 16 blocks x 4 waves
  k_gemm_wmma<HID, DIM, false><<<dim3(16), 128, 0, stream>>>(h1, W2, b2, out);
}